// SoftHistogram_4191888081298
// MI455X (gfx1250) — compile-verified
//
#include <hip/hip_runtime.h>

// SoftHistogram for MI455X (gfx1250, wave32).
// out[b, f*K+k] = (1/N) * sum_n relu( w[f,k] * |x[b,n,f] + c[f,k]| + b[f,k] )
// B=64, N=1000, F=256, K=8.
//
// Phase 1 (memory/VALU bound): stream x straight to VGPRs (no reuse -> bypass
//   LDS), 1 feature per thread, 8 bin accumulators in registers, N split into
//   8 chunks across blockIdx.y for occupancy. Partials -> d_ws.
// Phase 2 (tiny, matrix-unit): reduce the 8 chunk-partials per channel with
//   V_WMMA_F32_16X16X4_F32 against an all-(1/N) B matrix (K=4 contracts 4
//   chunks per WMMA; two WMMAs cover all 8 chunks). Deterministic, no atomics.

namespace {
constexpr int Kbins = 8;
constexpr int Fdim  = 256;
constexpr int Bdim  = 64;
constexpr int Ndim  = 1000;
constexpr int SCH   = 8;             // N-chunks (grid.y)
constexpr int NC    = Ndim / SCH;    // 125 rows per chunk
constexpr int CH    = Fdim * Kbins;  // 2048 output channels per batch
}

typedef float v2f __attribute__((ext_vector_type(2)));
typedef float v8f __attribute__((ext_vector_type(8)));

// ---------------- Phase 1: streaming partial accumulation --------------------
__global__ __launch_bounds__(256) void soft_hist_partial_kernel(
    const float* __restrict__ x,        // [B, N, F]
    const float* __restrict__ centers,  // [F*K]
    const float* __restrict__ widths_w, // [F*K]
    const float* __restrict__ widths_b, // [F*K]
    float* __restrict__ dst,            // mode 0: part [SCH][B][CH]; mode 1: out [B][CH]
    int atomic_mode)
{
  const int b   = blockIdx.x;
  const int s   = blockIdx.y;
  const int f   = threadIdx.x;     // one feature column per thread
  const int ch0 = f * Kbins;

  // Hoist per-channel affine params into registers (read once).
  float c[Kbins], w[Kbins], bb[Kbins];
#pragma unroll
  for (int k = 0; k < Kbins; ++k) {
    c[k]  = centers[ch0 + k];
    w[k]  = widths_w[ch0 + k];
    bb[k] = widths_b[ch0 + k];
  }

  float acc[Kbins];
#pragma unroll
  for (int k = 0; k < Kbins; ++k) acc[k] = 0.0f;

  // x[b, n, f]: f is contiguous -> lanes 0..31 read 128B contiguous per n.
  const float* __restrict__ xcol = x + (size_t)b * Ndim * Fdim + f;
  const int n0 = s * NC;

#pragma unroll 5
  for (int n = n0; n < n0 + NC; ++n) {
    // Single-use stream: non-temporal load (TH=NT), don't pollute L2.
    const float xv = __builtin_nontemporal_load(xcol + (size_t)n * Fdim);
#pragma unroll
    for (int k = 0; k < Kbins; ++k) {
      // v_add_f32 ; v_fma_f32 with |src0| modifier ; v_max_f32 ; v_add_f32
      const float r = fmaf(fabsf(xv + c[k]), w[k], bb[k]);
      acc[k] += fmaxf(r, 0.0f);
    }
  }

  if (atomic_mode == 0) {
    float4* p = (float4*)(dst + (size_t)(s * Bdim + b) * CH + ch0);
    p[0] = make_float4(acc[0], acc[1], acc[2], acc[3]);
    p[1] = make_float4(acc[4], acc[5], acc[6], acc[7]);
  } else {
    // Fallback path when d_ws is too small: accumulate into pre-zeroed out.
    const float invN = 1.0f / (float)Ndim;
    float* p = dst + (size_t)b * CH + ch0;
#pragma unroll
    for (int k = 0; k < Kbins; ++k) atomicAdd(p + k, acc[k] * invN);
  }
}

// ---------------- Phase 2: WMMA chunk reduction ------------------------------
// One wave reduces 16 consecutive channels for one batch b:
//   A (16x4 f32): A[m, kk] = part[s=kk][b][ch0+m]   (two WMMAs: s=0..3, 4..7)
//   B (4x16 f32): all elements = 1/N  (layout-invariant)
//   C (16x16 f32): every column j holds (1/N) * sum_s part[s][b][ch0+m]
// A layout (ISA 7.12.2): lanes 0-15 hold K=0,1 in VGPR0,1; lanes 16-31 hold K=2,3.
// C layout: VGPR v -> M=v (lanes 0-15) / M=v+8 (lanes 16-31); column = lane%16.
__global__ __launch_bounds__(256) void soft_hist_wmma_reduce_kernel(
    const float* __restrict__ part,  // [SCH][B][CH]
    float* __restrict__ out)         // [B][CH]
{
  const int wave = (blockIdx.x * blockDim.x + threadIdx.x) >> 5;
  const int lane = threadIdx.x & 31;
  const int b    = wave >> 7;        // 128 channel-groups (of 16) per batch
  const int g    = wave & 127;
  const int ch0  = g * 16;
  const int m    = lane & 15;        // A-matrix row owned by this lane
  const int half = lane >> 4;        // 0 -> K slots {0,1}; 1 -> K slots {2,3}

  const size_t sStride = (size_t)Bdim * CH;
  const size_t colOff  = (size_t)b * CH + ch0 + m;

  v2f a0, a1;
  a0.x = part[(size_t)(2 * half + 0) * sStride + colOff];  // s = 0..3
  a0.y = part[(size_t)(2 * half + 1) * sStride + colOff];
  a1.x = part[(size_t)(2 * half + 4) * sStride + colOff];  // s = 4..7
  a1.y = part[(size_t)(2 * half + 5) * sStride + colOff];

  const float invN = 1.0f / (float)Ndim;
  v2f ones; ones.x = invN; ones.y = invN;

  // EXEC is all-1s here (uniform control flow) as WMMA requires.
  v8f acc = {};
  acc = __builtin_amdgcn_wmma_f32_16x16x4_f32(false, a0, false, ones,
                                              (short)0, acc, false, false);
  acc = __builtin_amdgcn_wmma_f32_16x16x4_f32(false, a1, false, ones,
                                              (short)0, acc, false, false);

  // Column 0 lives in lanes 0 (M=0..7) and 16 (M=8..15).
  if (m == 0) {
    float* p = out + (size_t)b * CH + ch0 + 8 * half;
    ((float4*)p)[0] = make_float4(acc[0], acc[1], acc[2], acc[3]);
    ((float4*)p)[1] = make_float4(acc[4], acc[5], acc[6], acc[7]);
  }
}

// ---------------- Fallback zero kernel ---------------------------------------
__global__ void soft_hist_zero_kernel(float* p, int n) {
  const int i = blockIdx.x * blockDim.x + threadIdx.x;
  if (i < n) p[i] = 0.0f;
}

// ---------------- Launch ------------------------------------------------------
extern "C" void kernel_launch(void* const* d_in, const int* in_sizes, int n_in,
                              void* d_out, int out_size, void* d_ws, size_t ws_size,
                              hipStream_t stream) {
  const float* x  = (const float*)d_in[0];
  const float* c  = (const float*)d_in[1];
  const float* ww = (const float*)d_in[2];
  const float* wb = (const float*)d_in[3];
  float* out = (float*)d_out;

  const size_t ws_need = (size_t)SCH * Bdim * CH * sizeof(float);  // 4 MiB
  if (d_ws != nullptr && ws_size >= ws_need) {
    float* part = (float*)d_ws;
    dim3 grid(Bdim, SCH);  // 512 blocks x 8 waves = 4096 waves
    soft_hist_partial_kernel<<<grid, 256, 0, stream>>>(x, c, ww, wb, part, 0);

    const int waves  = Bdim * (CH / 16);  // 8192 wave-tasks
    const int blocks = waves / 8;         // 256 threads = 8 waves per block
    soft_hist_wmma_reduce_kernel<<<blocks, 256, 0, stream>>>(part, out);
  } else {
    const int n = Bdim * CH;
    soft_hist_zero_kernel<<<(n + 255) / 256, 256, 0, stream>>>(out, n);
    dim3 grid(Bdim, SCH);
    soft_hist_partial_kernel<<<grid, 256, 0, stream>>>(x, c, ww, wb, out, 1);
  }
}